// VRGCN_32684701122919
// MI455X (gfx1250) — compile-verified
//
#include <hip/hip_runtime.h>
#include <hip/hip_bf16.h>
#include <math.h>

typedef __attribute__((ext_vector_type(2))) float v2f;
typedef __attribute__((ext_vector_type(8))) float v8f;

#define D_DIM 512

// ---------------------------------------------------------------------------
// Zero-fill (accumulators must be re-zeroed every call: atomically accumulated)
// ---------------------------------------------------------------------------
__global__ __launch_bounds__(256)
void vrgcn_zero(float4* __restrict__ p, int n4) {
  int i = blockIdx.x * 256 + threadIdx.x;
  if (i < n4) p[i] = make_float4(0.f, 0.f, 0.f, 0.f);
}

// ---------------------------------------------------------------------------
// Edge-parallel SpMM with fused gather/subtract.
// out[rows[e]] += vals[e] * (srcA[idxA? idxA[cols[e]] : cols[e]]
//                            - (srcB? srcB[idxB? idxB[cols[e]] : cols[e]] : 0))
// One wave (32 lanes) per edge; lane covers 4 float4 chunks of the 512-wide row.
// ---------------------------------------------------------------------------
__global__ __launch_bounds__(256)
void vrgcn_spmm_edges(const int* __restrict__ rows, const int* __restrict__ cols,
                      const float* __restrict__ vals, int nE,
                      const float* __restrict__ srcA, const int* __restrict__ idxA,
                      const float* __restrict__ srcB, const int* __restrict__ idxB,
                      float* __restrict__ out)
{
  const int e = blockIdx.x * 8 + (threadIdx.x >> 5);
  if (e >= nE) return;
  const int lane = threadIdx.x & 31;
  const int r = rows[e];
  const int c = cols[e];
  const float v = vals[e];

  const int ia = idxA ? idxA[c] : c;
  const float4* pa = (const float4*)(srcA + (size_t)ia * D_DIM);
  const float4* pb = nullptr;
  if (srcB) {
    const int ib = idxB ? idxB[c] : c;
    pb = (const float4*)(srcB + (size_t)ib * D_DIM);
  }
  float* dst = out + (size_t)r * D_DIM;

#pragma unroll
  for (int i = 0; i < 4; ++i) {
    const int q = lane + i * 32;          // float4 index 0..127, coalesced
    float4 t = pa[q];
    if (pb) {
      float4 s = pb[q];
      t.x -= s.x; t.y -= s.y; t.z -= s.z; t.w -= s.w;
    }
    atomicAdd(dst + q * 4 + 0, v * t.x);
    atomicAdd(dst + q * 4 + 1, v * t.y);
    atomicAdd(dst + q * 4 + 2, v * t.z);
    atomicAdd(dst + q * 4 + 3, v * t.w);
  }
}

// ---------------------------------------------------------------------------
// C[M,512] = A[M,512] @ W[512,512]^T + bias, fp32 WMMA 16x16x4.
// Block: 64x64 output tile, 256 threads = 8 waves (4 M-strips x 2 N-cols).
// Each wave: one 16-row strip, two 16x16 accumulators (32 N-cols).
// LDS tiles stored transposed [k][m] with +1 padding -> conflict-free
// fragment reads matching the ISA A/B VGPR layout.
// ---------------------------------------------------------------------------
#define TILE_K 32

__global__ __launch_bounds__(256)
void vrgcn_gemm_bias(const float* __restrict__ A, const float* __restrict__ W,
                     const float* __restrict__ bias, float* __restrict__ C, int M)
{
  __shared__ float As[TILE_K][64 + 1];
  __shared__ float Bs[TILE_K][64 + 1];

  const int tid  = threadIdx.x;
  const int lane = tid & 31;
  const int wave = tid >> 5;
  const int wm   = wave & 3;        // M strip 0..3
  const int wn   = wave >> 2;       // N half 0..1

  const int blockM = blockIdx.x * 64;
  const int blockN = blockIdx.y * 64;

  // global->LDS mapping: thread loads float4 at (row = tid>>3 [+32], col = (tid&7)*4)
  const int lrow = tid >> 3;        // 0..31
  const int lk   = (tid & 7) * 4;   // 0..28

  const float* aPtr = A + (size_t)(blockM + lrow) * D_DIM + lk;
  const float* wPtr = W + (size_t)(blockN + lrow) * D_DIM + lk;

  v8f acc0 = {};
  v8f acc1 = {};

  // WMMA fragment indices (ISA: A lanes 0-15 hold M rows, VGPR half = K pair)
  const int mFrag = wm * 16 + (lane & 15);
  const int nFrag = wn * 32 + (lane & 15);
  const int kHalf = (lane >> 4) * 2;

  for (int k0 = 0; k0 < D_DIM; k0 += TILE_K) {
    float4 a0 = *(const float4*)(aPtr + k0);
    float4 a1 = *(const float4*)(aPtr + k0 + 32 * D_DIM);
    float4 b0 = *(const float4*)(wPtr + k0);
    float4 b1 = *(const float4*)(wPtr + k0 + 32 * D_DIM);

    __syncthreads();
    As[lk + 0][lrow]      = a0.x; As[lk + 1][lrow]      = a0.y;
    As[lk + 2][lrow]      = a0.z; As[lk + 3][lrow]      = a0.w;
    As[lk + 0][lrow + 32] = a1.x; As[lk + 1][lrow + 32] = a1.y;
    As[lk + 2][lrow + 32] = a1.z; As[lk + 3][lrow + 32] = a1.w;
    Bs[lk + 0][lrow]      = b0.x; Bs[lk + 1][lrow]      = b0.y;
    Bs[lk + 2][lrow]      = b0.z; Bs[lk + 3][lrow]      = b0.w;
    Bs[lk + 0][lrow + 32] = b1.x; Bs[lk + 1][lrow + 32] = b1.y;
    Bs[lk + 2][lrow + 32] = b1.z; Bs[lk + 3][lrow + 32] = b1.w;
    __syncthreads();

#pragma unroll
    for (int kk = 0; kk < TILE_K; kk += 4) {
      v2f af, bf0, bf1;
      af.x  = As[kk + kHalf + 0][mFrag];
      af.y  = As[kk + kHalf + 1][mFrag];
      bf0.x = Bs[kk + kHalf + 0][nFrag];
      bf0.y = Bs[kk + kHalf + 1][nFrag];
      bf1.x = Bs[kk + kHalf + 0][nFrag + 16];
      bf1.y = Bs[kk + kHalf + 1][nFrag + 16];
      acc0 = __builtin_amdgcn_wmma_f32_16x16x4_f32(false, af, false, bf0,
                                                   (short)0, acc0, false, false);
      acc1 = __builtin_amdgcn_wmma_f32_16x16x4_f32(false, af, false, bf1,
                                                   (short)0, acc1, false, false);
    }
  }

  // C/D layout: VGPR j holds rows {j, j+8} selected by lane>>4, cols = lane&15
  const int cN0 = blockN + wn * 32 + (lane & 15);
  const int cN1 = cN0 + 16;
  const float bias0 = bias[cN0];
  const float bias1 = bias[cN1];
  const int cM = blockM + wm * 16 + (lane >> 4) * 8;
#pragma unroll
  for (int j = 0; j < 8; ++j) {
    C[(size_t)(cM + j) * D_DIM + cN0] = acc0[j] + bias0;
    C[(size_t)(cM + j) * D_DIM + cN1] = acc1[j] + bias1;
  }
}

// ---------------------------------------------------------------------------
// relu(layernorm(x)) — one wave per row; lane holds 16 floats in registers.
// Safe in-place (full row read before write).
// ---------------------------------------------------------------------------
__global__ __launch_bounds__(256)
void vrgcn_ln_relu(const float* __restrict__ in, const float* __restrict__ g,
                   const float* __restrict__ beta, float* __restrict__ out, int M)
{
  const int row = blockIdx.x * 8 + (threadIdx.x >> 5);
  if (row >= M) return;
  const int lane = threadIdx.x & 31;

  const float4* p = (const float4*)(in + (size_t)row * D_DIM);
  float4 vv[4];
  float s = 0.f;
#pragma unroll
  for (int i = 0; i < 4; ++i) {
    vv[i] = p[lane + 32 * i];
    s += vv[i].x + vv[i].y + vv[i].z + vv[i].w;
  }
#pragma unroll
  for (int off = 16; off > 0; off >>= 1) s += __shfl_xor(s, off, 32);
  const float mu = s * (1.0f / 512.0f);

  float vs = 0.f;
#pragma unroll
  for (int i = 0; i < 4; ++i) {
    float dx = vv[i].x - mu, dy = vv[i].y - mu, dz = vv[i].z - mu, dw = vv[i].w - mu;
    vs += dx * dx + dy * dy + dz * dz + dw * dw;
  }
#pragma unroll
  for (int off = 16; off > 0; off >>= 1) vs += __shfl_xor(vs, off, 32);
  const float inv = rsqrtf(vs * (1.0f / 512.0f) + 1e-5f);

  const float4* gp = (const float4*)g;
  const float4* bp = (const float4*)beta;
  float4* o = (float4*)(out + (size_t)row * D_DIM);
#pragma unroll
  for (int i = 0; i < 4; ++i) {
    const int q = lane + 32 * i;
    float4 gg = gp[q], bb = bp[q], r;
    r.x = fmaxf(0.f, (vv[i].x - mu) * inv * gg.x + bb.x);
    r.y = fmaxf(0.f, (vv[i].y - mu) * inv * gg.y + bb.y);
    r.z = fmaxf(0.f, (vv[i].z - mu) * inv * gg.z + bb.z);
    r.w = fmaxf(0.f, (vv[i].w - mu) * inv * gg.w + bb.w);
    o[q] = r;
  }
}

// ---------------------------------------------------------------------------
// log_softmax over rows of [M,512], in place; one wave per row.
// ---------------------------------------------------------------------------
__global__ __launch_bounds__(256)
void vrgcn_log_softmax(float* __restrict__ io, int M)
{
  const int row = blockIdx.x * 8 + (threadIdx.x >> 5);
  if (row >= M) return;
  const int lane = threadIdx.x & 31;

  float4* p = (float4*)(io + (size_t)row * D_DIM);
  float4 vv[4];
  float mx = -INFINITY;
#pragma unroll
  for (int i = 0; i < 4; ++i) {
    vv[i] = p[lane + 32 * i];
    mx = fmaxf(mx, fmaxf(fmaxf(vv[i].x, vv[i].y), fmaxf(vv[i].z, vv[i].w)));
  }
#pragma unroll
  for (int off = 16; off > 0; off >>= 1) mx = fmaxf(mx, __shfl_xor(mx, off, 32));

  float s = 0.f;
#pragma unroll
  for (int i = 0; i < 4; ++i) {
    s += expf(vv[i].x - mx) + expf(vv[i].y - mx) +
         expf(vv[i].z - mx) + expf(vv[i].w - mx);
  }
#pragma unroll
  for (int off = 16; off > 0; off >>= 1) s += __shfl_xor(s, off, 32);
  const float lse = mx + logf(s);

#pragma unroll
  for (int i = 0; i < 4; ++i) {
    float4 r;
    r.x = vv[i].x - lse; r.y = vv[i].y - lse;
    r.z = vv[i].z - lse; r.w = vv[i].w - lse;
    p[lane + 32 * i] = r;
  }
}

// ---------------------------------------------------------------------------
// Launch
// ---------------------------------------------------------------------------
extern "C" void kernel_launch(void* const* d_in, const int* in_sizes, int n_in,
                              void* d_out, int out_size, void* d_ws, size_t ws_size,
                              hipStream_t stream) {
  const float* x          = (const float*)d_in[0];
  const float* hist0      = (const float*)d_in[1];
  const float* hist1      = (const float*)d_in[2];
  const float* w0         = (const float*)d_in[3];
  const float* b0         = (const float*)d_in[4];
  const float* g0         = (const float*)d_in[5];
  const float* beta0      = (const float*)d_in[6];
  const float* w1         = (const float*)d_in[7];
  const float* b1         = (const float*)d_in[8];
  const float* sadj0_vals = (const float*)d_in[9];
  const float* sadj1_vals = (const float*)d_in[10];
  const float* fadj0_vals = (const float*)d_in[11];
  const float* fadj1_vals = (const float*)d_in[12];
  const int*   sid0       = (const int*)d_in[13];
  const int*   sid1       = (const int*)d_in[14];
  /* sid2 = d_in[15] only defines N2 */
  const int*   fid0       = (const int*)d_in[16];
  const int*   fid1       = (const int*)d_in[17];
  const int*   sadj0_rows = (const int*)d_in[18];
  const int*   sadj0_cols = (const int*)d_in[19];
  const int*   sadj1_rows = (const int*)d_in[20];
  const int*   sadj1_cols = (const int*)d_in[21];
  const int*   fadj0_rows = (const int*)d_in[22];
  const int*   fadj0_cols = (const int*)d_in[23];
  const int*   fadj1_rows = (const int*)d_in[24];
  const int*   fadj1_cols = (const int*)d_in[25];

  const int N1  = in_sizes[14];          // 32768
  const int N2  = in_sizes[15];          // 8192
  const int Es0 = in_sizes[9];
  const int Es1 = in_sizes[10];
  const int Ef0 = in_sizes[11];
  const int Ef1 = in_sizes[12];

  float* acc1 = (float*)d_ws;                         // [N1,512] spmm accum L0
  float* hbuf = acc1 + (size_t)N1 * D_DIM;            // [N1,512] gemm0 / LN out
  float* acc2 = hbuf + (size_t)N1 * D_DIM;            // [N2,512] spmm accum L1
  float* out  = (float*)d_out;                        // [N2,512]

  // zero accumulators (required every call)
  {
    int n4 = N1 * D_DIM / 4;
    vrgcn_zero<<<(n4 + 255) / 256, 256, 0, stream>>>((float4*)acc1, n4);
    n4 = N2 * D_DIM / 4;
    vrgcn_zero<<<(n4 + 255) / 256, 256, 0, stream>>>((float4*)acc2, n4);
  }

  // layer 0: spmm(sadj0, x[sid0]-hist0[sid0]) + spmm(fadj0, hist0[fid0])
  vrgcn_spmm_edges<<<(Es0 + 7) / 8, 256, 0, stream>>>(
      sadj0_rows, sadj0_cols, sadj0_vals, Es0, x, sid0, hist0, sid0, acc1);
  vrgcn_spmm_edges<<<(Ef0 + 7) / 8, 256, 0, stream>>>(
      fadj0_rows, fadj0_cols, fadj0_vals, Ef0, hist0, fid0, nullptr, nullptr, acc1);

  // h = relu(LN(acc1 @ w0^T + b0))
  {
    dim3 grid(N1 / 64, D_DIM / 64);
    vrgcn_gemm_bias<<<grid, 256, 0, stream>>>(acc1, w0, b0, hbuf, N1);
  }
  vrgcn_ln_relu<<<(N1 + 7) / 8, 256, 0, stream>>>(hbuf, g0, beta0, hbuf, N1);

  // layer 1: spmm(sadj1, h - hist1[sid1]) + spmm(fadj1, hist1[fid1])
  vrgcn_spmm_edges<<<(Es1 + 7) / 8, 256, 0, stream>>>(
      sadj1_rows, sadj1_cols, sadj1_vals, Es1, hbuf, nullptr, hist1, sid1, acc2);
  vrgcn_spmm_edges<<<(Ef1 + 7) / 8, 256, 0, stream>>>(
      fadj1_rows, fadj1_cols, fadj1_vals, Ef1, hist1, fid1, nullptr, nullptr, acc2);

  // out = log_softmax(acc2 @ w1^T + b1)
  {
    dim3 grid(N2 / 64, D_DIM / 64);
    vrgcn_gemm_bias<<<grid, 256, 0, stream>>>(acc2, w1, b1, out, N2);
  }
  vrgcn_log_softmax<<<(N2 + 7) / 8, 256, 0, stream>>>(out, N2);

  (void)n_in; (void)out_size; (void)ws_size;
}